// CPSFMemcellFusedReal_12051678233212
// MI455X (gfx1250) — compile-verified
//
#include <hip/hip_runtime.h>
#include <cfloat>
#include <cmath>

// Problem sizes (fixed by the reference)
#define B_ 2048
#define M_ 2048
#define N_ 32
#define S_ 256

// K-panel chunking for the K=2048 GEMMs
#define KC_  64          // K elements staged per chunk
#define KCP_ (KC_ + 2)   // padded LDS row stride (floats): 66 -> bank stride 2

static constexpr float EPS_DIR = 1e-6f;
static constexpr float MAXQ    = 25.0f;
static constexpr float CAP_    = 1.0f;
static constexpr float PI_F    = 3.14159265358979323846f;
static constexpr float F32EPS  = 1.1920928955078125e-7f;   // FLT_EPSILON
static constexpr float TINY_   = 1.1754943508222875e-38f;  // FLT_MIN (jnp tiny)

typedef __attribute__((ext_vector_type(2))) float v2f;
typedef __attribute__((ext_vector_type(8))) float v8f;

__device__ __forceinline__ float softplusf(float x) {
    return fmaxf(x, 0.0f) + log1pf(expf(-fabsf(x)));  // stable log1p(exp(x))
}

__device__ __forceinline__ float wave_sum32(float x) {
#pragma unroll
    for (int o = 16; o > 0; o >>= 1) x += __shfl_xor(x, o, 32);
    return x;
}

// ---------------------------------------------------------------------------
// Async (ASYNCcnt-tracked) global->LDS panel copy with on-the-fly transpose.
//   Copies src[k0:k0+KC_, s0:s0+32] (row stride ld) into LDS laid out as
//   panel[s][k] with padded stride KCP_.  Per-lane scatter: the async-to-LDS
//   instruction takes an independent LDS byte address per lane, so the
//   transpose costs nothing.  GVS addressing: saddr(base) + vgpr(offset).
// ---------------------------------------------------------------------------
__device__ __forceinline__ void async_panel_load(float* lds_base,
                                                 const float* __restrict__ gsrc,
                                                 int k0, int s0, int ld, int tid) {
#pragma unroll
    for (int i = 0; i < (KC_ * 32) / 256; ++i) {     // 8 B32 transfers / thread
        int e = i * 256 + tid;
        int s = e & 31;                               // 16 consecutive lanes ->
        int k = e >> 5;                               // 64B contiguous global
        uint32_t lds  = (uint32_t)(uintptr_t)(lds_base + s * KCP_ + k);
        uint32_t gofs = (uint32_t)(((k0 + k) * ld + s0 + s) * sizeof(float));
        asm volatile("global_load_async_to_lds_b32 %0, %1, %2"
                     :: "v"(lds), "v"(gofs), "s"(gsrc) : "memory");
    }
}

__device__ __forceinline__ void wait_async0() {
    asm volatile("s_wait_asynccnt 0x0" ::: "memory");
}

// ---------------------------------------------------------------------------
// Kernel 1: per-row precompute (one wave32 per row; N == 32).
// ---------------------------------------------------------------------------
__global__ void precompute_kernel(const float* __restrict__ z,
                                  const float* __restrict__ zj,
                                  const float* __restrict__ vd,
                                  const float* __restrict__ sig_par,
                                  const float* __restrict__ sig_perp,
                                  float* __restrict__ bdir,
                                  float* __restrict__ wperp,
                                  float* __restrict__ wdiff,
                                  float* __restrict__ c1,
                                  float* __restrict__ c2,
                                  float* __restrict__ znorm) {
    const int lane = threadIdx.x & 31;
    const int row  = blockIdx.x * (blockDim.x >> 5) + (threadIdx.x >> 5);
    if (row < M_) {
        float v  = vd[row * N_ + lane];
        float zv = zj[row * N_ + lane];
        float dn = sqrtf(wave_sum32(v * v));
        float bv = (dn > EPS_DIR) ? (v / dn) : 0.0f;  // mask folded into b_dir
        bdir[row * N_ + lane] = bv;
        float s1 = wave_sum32(zv * zv);
        float s2 = wave_sum32(zv * bv);
        if (lane == 0) {
            c1[row] = s1;
            c2[row] = s2;
            float sp_par  = fmaxf(softplusf(sig_par[row])  + F32EPS, F32EPS);
            float sp_perp = fmaxf(softplusf(sig_perp[row]) + F32EPS, F32EPS);
            float wpar    = 1.0f / (sp_par * sp_par);
            float wperp_v = 1.0f / (sp_perp * sp_perp);
            wperp[row] = wperp_v;
            wdiff[row] = wpar - wperp_v;
        }
    }
    if (row < B_) {
        float zv = z[row * N_ + lane];
        float s  = wave_sum32(zv * zv);
        if (lane == 0) znorm[row] = s;
    }
}

// ---------------------------------------------------------------------------
// Kernel 2: T_hat_eff = T_hat_j + T_hat_j_delta
// ---------------------------------------------------------------------------
__global__ void theff_kernel(const float* __restrict__ That,
                             const float* __restrict__ Delta,
                             float* __restrict__ theff) {
    for (int i = blockIdx.x * blockDim.x + threadIdx.x; i < M_ * S_;
         i += gridDim.x * blockDim.x)
        theff[i] = That[i] + Delta[i];
}

// ---------------------------------------------------------------------------
// Kernel 3: gain[B,M] via two fused K=32 GEMMs on V_WMMA_F32_16X16X4_F32
//   plus the anisotropic-quadratic / softplus-clamp / exp epilogue.
// ---------------------------------------------------------------------------
__global__ void gain_kernel(const float* __restrict__ z,
                            const float* __restrict__ zj,
                            const float* __restrict__ bdir,
                            const float* __restrict__ alpha_j,
                            const float* __restrict__ wperp,
                            const float* __restrict__ wdiff,
                            const float* __restrict__ c1,
                            const float* __restrict__ c2,
                            const float* __restrict__ znorm,
                            float* __restrict__ gain) {
    const int lane   = threadIdx.x & 31;
    const int tile   = blockIdx.x * (blockDim.x >> 5) + (threadIdx.x >> 5);
    const int tilesM = M_ / 16;
    const int b0 = (tile / tilesM) * 16;
    const int m0 = (tile % tilesM) * 16;
    const int row = lane & 15;
    const int kk  = (lane >> 4) << 1;

    v8f acc_zz = {};
    v8f acc_zb = {};
#pragma unroll
    for (int k0 = 0; k0 < N_; k0 += 4) {
        const int kb = k0 + kk;
        v2f a  = *(const v2f*)(z    + (size_t)(b0 + row) * N_ + kb);
        v2f bz = *(const v2f*)(zj   + (size_t)(m0 + row) * N_ + kb);
        v2f bd = *(const v2f*)(bdir + (size_t)(m0 + row) * N_ + kb);
        acc_zz = __builtin_amdgcn_wmma_f32_16x16x4_f32(false, a, false, bz,
                                                       (short)0, acc_zz, false, false);
        acc_zb = __builtin_amdgcn_wmma_f32_16x16x4_f32(false, a, false, bd,
                                                       (short)0, acc_zb, false, false);
    }

    const int   mcol = m0 + row;
    const float wp  = wperp[mcol];
    const float wd  = wdiff[mcol];
    const float cc1 = c1[mcol];
    const float cc2 = c2[mcol];
    const float al  = alpha_j[mcol];
    const int   hi  = lane >> 4;
#pragma unroll
    for (int r = 0; r < 8; ++r) {
        const int brow = b0 + r + 8 * hi;
        float dzsq = znorm[brow] - 2.0f * acc_zz[r] + cc1;
        float pr   = acc_zb[r] - cc2;
        float q    = wp * dzsq + wd * pr * pr;
        q = MAXQ - softplusf(MAXQ - q);
        gain[(size_t)brow * M_ + mcol] = al * expf(-PI_F * q);
    }
}

// ---------------------------------------------------------------------------
// Kernel 4 / 8: out[B,S] = gain[B,M] @ X[M,S], K = M = 2048.
//   Block tile 128(B) x 32(S): 8 waves, each owns 16 rows and both 16-col
//   halves (2 accumulators).  X K-panel (KC_ x 32) is async-copied to LDS
//   (transposed, double-buffered) and shared by all 8 waves; B fragments are
//   single conflict-free ds_load_b64s.  mode 1: out = acc - T_star (E_eff).
// ---------------------------------------------------------------------------
__global__ void gemm_gainX_kernel(const float* __restrict__ gain,
                                  const float* __restrict__ X,
                                  const float* __restrict__ Tstar,
                                  float* __restrict__ out,
                                  int subtract_tstar) {
    __shared__ float panel[2][32 * KCP_];
    const int lane = threadIdx.x & 31;
    const int wave = threadIdx.x >> 5;
    const int tid  = threadIdx.x;
    const int b0 = (blockIdx.x >> 3) * 128 + wave * 16;   // 16 B-blocks
    const int s0 = (blockIdx.x & 7) * 32;                 // 8 S-blocks
    const int row = lane & 15;
    const int kk  = (lane >> 4) << 1;

    async_panel_load(panel[0], X, 0, s0, S_, tid);
    wait_async0();
    __syncthreads();

    v8f acc0 = {};
    v8f acc1 = {};
    const int nk = M_ / KC_;          // 32 chunks
    for (int c = 0; c < nk; ++c) {
        const int cur = c & 1;
        if (c + 1 < nk)
            async_panel_load(panel[cur ^ 1], X, (c + 1) * KC_, s0, S_, tid);

        const float* gA  = gain + (size_t)(b0 + row) * M_ + c * KC_ + kk;
        const float* pB0 = &panel[cur][row * KCP_];
        const float* pB1 = &panel[cur][(row + 16) * KCP_];
#pragma unroll
        for (int k = 0; k < KC_; k += 4) {
            v2f a  = *(const v2f*)(gA + k);
            v2f b0 = *(const v2f*)(pB0 + k + kk);
            v2f b1 = *(const v2f*)(pB1 + k + kk);
            acc0 = __builtin_amdgcn_wmma_f32_16x16x4_f32(false, a, false, b0,
                                                         (short)0, acc0, false, false);
            acc1 = __builtin_amdgcn_wmma_f32_16x16x4_f32(false, a, false, b1,
                                                         (short)0, acc1, false, false);
        }
        if (c + 1 < nk) {
            wait_async0();
            __syncthreads();
        }
    }

    const int hi = lane >> 4;
#pragma unroll
    for (int r = 0; r < 8; ++r) {
        const int orow = b0 + r + 8 * hi;
        float v0 = acc0[r];
        float v1 = acc1[r];
        if (subtract_tstar) {
            v0 -= Tstar[(size_t)orow * S_ + s0 + row];
            v1 -= Tstar[(size_t)orow * S_ + s0 + 16 + row];
        }
        out[(size_t)orow * S_ + s0 + row]      = v0;
        out[(size_t)orow * S_ + s0 + 16 + row] = v1;
    }
}

// ---------------------------------------------------------------------------
// Kernel 5: dpre[M,S] = (-sigmoid(alpha_logit)/B) * (gain^T @ E), K = B.
//   Same 128x32 block structure; E K-panel staged async/double-buffered in
//   LDS; A (gain transposed) via coalesced 16-lane global loads.
// ---------------------------------------------------------------------------
__global__ void grad_kernel(const float* __restrict__ gain,
                            const float* __restrict__ E,
                            const float* __restrict__ alogit,
                            float* __restrict__ dpre) {
    __shared__ float panel[2][32 * KCP_];
    const int lane = threadIdx.x & 31;
    const int wave = threadIdx.x >> 5;
    const int tid  = threadIdx.x;
    const int m0 = (blockIdx.x >> 3) * 128 + wave * 16;   // 16 M-blocks
    const int s0 = (blockIdx.x & 7) * 32;                 // 8 S-blocks
    const int row = lane & 15;
    const int kk  = (lane >> 4) << 1;

    async_panel_load(panel[0], E, 0, s0, S_, tid);
    wait_async0();
    __syncthreads();

    v8f acc0 = {};
    v8f acc1 = {};
    const int nk = B_ / KC_;          // 32 chunks
    for (int c = 0; c < nk; ++c) {
        const int cur = c & 1;
        if (c + 1 < nk)
            async_panel_load(panel[cur ^ 1], E, (c + 1) * KC_, s0, S_, tid);

        const int k0 = c * KC_;
        const float* pB0 = &panel[cur][row * KCP_];
        const float* pB1 = &panel[cur][(row + 16) * KCP_];
#pragma unroll
        for (int k = 0; k < KC_; k += 4) {
            v2f a;   // A[m, k=b] = gain[b, m]  (transposed access)
            a.x = gain[(size_t)(k0 + k + kk)     * M_ + m0 + row];
            a.y = gain[(size_t)(k0 + k + kk + 1) * M_ + m0 + row];
            v2f b0 = *(const v2f*)(pB0 + k + kk);
            v2f b1 = *(const v2f*)(pB1 + k + kk);
            acc0 = __builtin_amdgcn_wmma_f32_16x16x4_f32(false, a, false, b0,
                                                         (short)0, acc0, false, false);
            acc1 = __builtin_amdgcn_wmma_f32_16x16x4_f32(false, a, false, b1,
                                                         (short)0, acc1, false, false);
        }
        if (c + 1 < nk) {
            wait_async0();
            __syncthreads();
        }
    }

    const float alpha = 1.0f / (1.0f + expf(-alogit[0]));
    const float coef  = -alpha / (float)B_;
    const int hi = lane >> 4;
#pragma unroll
    for (int r = 0; r < 8; ++r) {
        const int orow = m0 + r + 8 * hi;
        dpre[(size_t)orow * S_ + s0 + row]      = coef * acc0[r];
        dpre[(size_t)orow * S_ + s0 + 16 + row] = coef * acc1[r];
    }
}

// ---------------------------------------------------------------------------
// Kernel 6: deterministic single-block sum of squares -> sumsq[0].
// ---------------------------------------------------------------------------
__global__ void sumsq_kernel(const float* __restrict__ dpre,
                             float* __restrict__ sumsq) {
    __shared__ float red[32];
    float acc = 0.0f;
    for (int i = threadIdx.x; i < M_ * S_; i += blockDim.x) {
        float v = dpre[i];
        acc += v * v;
    }
    acc = wave_sum32(acc);
    const int lane = threadIdx.x & 31;
    const int wave = threadIdx.x >> 5;
    if (lane == 0) red[wave] = acc;
    __syncthreads();
    if (wave == 0) {
        float v = red[lane];          // blockDim == 1024 -> exactly 32 waves
        v = wave_sum32(v);
        if (lane == 0) sumsq[0] = v;
    }
}

// ---------------------------------------------------------------------------
// Kernel 7: W = T_hat_eff + s * dpre, s = min(CAP/(||dpre||_F + tiny), 1)
// ---------------------------------------------------------------------------
__global__ void wbuild_kernel(const float* __restrict__ theff,
                              const float* __restrict__ dpre,
                              const float* __restrict__ sumsq,
                              float* __restrict__ W) {
    const float n = sqrtf(sumsq[0]);
    const float s = fminf(CAP_ / (n + TINY_), 1.0f);
    for (int i = blockIdx.x * blockDim.x + threadIdx.x; i < M_ * S_;
         i += gridDim.x * blockDim.x)
        W[i] = theff[i] + s * dpre[i];
}

// ---------------------------------------------------------------------------
extern "C" void kernel_launch(void* const* d_in, const int* in_sizes, int n_in,
                              void* d_out, int out_size, void* d_ws, size_t ws_size,
                              hipStream_t stream) {
    (void)in_sizes; (void)n_in; (void)out_size; (void)ws_size;
    const float* z       = (const float*)d_in[0];
    const float* Tstar   = (const float*)d_in[1];
    const float* zj      = (const float*)d_in[2];
    const float* vd      = (const float*)d_in[3];
    const float* That    = (const float*)d_in[4];
    const float* Delta   = (const float*)d_in[5];
    const float* alphaj  = (const float*)d_in[6];
    const float* sigpar  = (const float*)d_in[7];
    const float* sigperp = (const float*)d_in[8];
    const float* alogit  = (const float*)d_in[9];

    float* ws = (float*)d_ws;
    size_t off = 0;
    float* bdir  = ws + off; off += (size_t)M_ * N_;   // 256 KB
    float* wperp = ws + off; off += M_;
    float* wdiff = ws + off; off += M_;
    float* c1    = ws + off; off += M_;
    float* c2    = ws + off; off += M_;
    float* znorm = ws + off; off += B_;
    float* gain  = ws + off; off += (size_t)B_ * M_;   // 16 MB
    float* theff = ws + off; off += (size_t)M_ * S_;   // 2 MB
    float* E     = ws + off; off += (size_t)B_ * S_;   // 2 MB
    float* dpre  = ws + off; off += (size_t)M_ * S_;   // 2 MB
    float* W     = ws + off; off += (size_t)M_ * S_;   // 2 MB
    float* sumsq = ws + off; off += 1;

    // 1) per-row constants (one wave32 per row; 8 rows per 256-thread block)
    precompute_kernel<<<256, 256, 0, stream>>>(z, zj, vd, sigpar, sigperp,
                                               bdir, wperp, wdiff, c1, c2, znorm);
    // 2) T_hat_eff
    theff_kernel<<<512, 256, 0, stream>>>(That, Delta, theff);
    // 3) gain[B,M]: (B/16)*(M/16)=16384 tiles, 8 waves/block
    gain_kernel<<<2048, 256, 0, stream>>>(z, zj, bdir, alphaj, wperp, wdiff,
                                          c1, c2, znorm, gain);
    // 4) E = gain @ T_hat_eff - T_star : 16 x 8 block tiles of 128x32
    gemm_gainX_kernel<<<128, 256, 0, stream>>>(gain, theff, Tstar, E, 1);
    // 5) dpre = -(sigmoid(alogit)/B) * gain^T @ E : 16 x 8 block tiles
    grad_kernel<<<128, 256, 0, stream>>>(gain, E, alogit, dpre);
    // 6) ||dpre||^2 (deterministic single-block reduction)
    sumsq_kernel<<<1, 1024, 0, stream>>>(dpre, sumsq);
    // 7) W = T_hat_eff + s * dpre
    wbuild_kernel<<<512, 256, 0, stream>>>(theff, dpre, sumsq, W);
    // 8) T = gain @ W  -> d_out
    gemm_gainX_kernel<<<128, 256, 0, stream>>>(gain, W, Tstar, (float*)d_out, 0);
}